// Mean_layer_11751030521986
// MI455X (gfx1250) — compile-verified
//
#include <hip/hip_runtime.h>
#include <hip/hip_bf16.h>

typedef float v2f __attribute__((ext_vector_type(2)));
typedef float v4f __attribute__((ext_vector_type(4)));
typedef float v8f __attribute__((ext_vector_type(8)));

#define WMMA_F32(A, B, C) \
    __builtin_amdgcn_wmma_f32_16x16x4_f32(false, (A), false, (B), (short)0, (C), false, false)

#define LOQ(Q) __builtin_shufflevector((Q), (Q), 0, 1)
#define HIQ(Q) __builtin_shufflevector((Q), (Q), 2, 3)

// ---------------------------------------------------------------------------
// Kernel 1: combined[b, 0:F]  = features_ue[nodes[b]]
//           combined[b, F:2F] = (sum_s features_ue[neigh_idx[b,s]]) / (S+1)
// ---------------------------------------------------------------------------
__global__ void gather_combine_kernel(const float* __restrict__ feat,
                                      const int* __restrict__ neigh_idx,
                                      const int* __restrict__ nodes,
                                      float* __restrict__ combined,
                                      int F, int S) {
    const int b = blockIdx.x;
    const int K = 2 * F;
    const int self_row = nodes[b];                 // uniform -> scalar load
    const float inv = 1.0f / (float)(S + 1);

    for (int f = threadIdx.x; f < F; f += blockDim.x) {
        float sf = feat[(size_t)self_row * F + f];
        float acc = 0.0f;
        for (int s = 0; s < S; ++s) {
            int id = neigh_idx[b * S + s];         // uniform per block
            acc += feat[(size_t)id * F + f];
        }
        combined[(size_t)b * K + f]     = sf;
        combined[(size_t)b * K + F + f] = acc * inv;
    }
}

// ---------------------------------------------------------------------------
// Kernel 2 (K compile-time): out[e,b] = relu( sum_k W[e,k] * combined[b,k] )
//   Block = 8 waves (256 thr), covers 256(M) x 32(N). Wave owns 32x32:
//   2 M-tiles x 2 N-tiles = 4 accumulators (low pressure -> in-place D=C).
//   K-permutation: lane-half h covers K {k0+4h, k0+4h+1} (lo) and
//   {k0+4h+2, k0+4h+3} (hi) — valid since WMMA pairs A/B positionally —
//   so one b128 load per row feeds two WMMA K-steps. Ping-pong q/p groups.
// ---------------------------------------------------------------------------
template <int K>
__global__ __launch_bounds__(256)
void wmma_gemm_relu_kernel(const float* __restrict__ W,
                           const float* __restrict__ combined,
                           float* __restrict__ out,
                           int Bn) {
    constexpr int KP = K + 4;               // row pad (floats): keeps 16B align
    __shared__ float lds[32 * KP];          // 32 combined rows, padded

    const int tid  = threadIdx.x;
    const int wave = tid >> 5;              // 0..7
    const int lane = tid & 31;
    const int r    = lane & 15;             // 0..15
    const int half = lane >> 4;             // 0 or 1

    const int b0 = blockIdx.x * 32;         // N tile base
    const int m0 = wave * 32;               // M strip base (2 tiles)

    // --- stage 32 contiguous combined rows into LDS ---
    {
        const float4* src = (const float4*)(combined + (size_t)b0 * K);
        constexpr int kv = K >> 2;          // float4s per row
        #pragma unroll 2
        for (int i = tid; i < 32 * kv; i += 256) {
            int row = i / kv;
            int col = i - row * kv;
            *(float4*)&lds[row * KP + (col << 2)] = src[i];
        }
    }
    __syncthreads();

    v8f c00 = {}, c01 = {}, c10 = {}, c11 = {};

    // Lane-varying bases; M-tile delta (16*K floats) becomes an immediate.
    const float* wb = W + (m0 + r) * K + 4 * half;
    const float* l0 = &lds[r        * KP + 4 * half];
    const float* l1 = &lds[(16 + r) * KP + 4 * half];

    v4f qA0, qA1, qf0, qf1;                 // group at k0
    v4f pA0, pA1, pf0, pf1;                 // group at k0+8

#define LOAD_GROUP(A0, A1, F0, F1, koff)                    \
    do {                                                    \
        A0 = *(const v4f*)&wb[0 * 16 * K + (koff)];         \
        A1 = *(const v4f*)&wb[1 * 16 * K + (koff)];         \
        F0 = *(const v4f*)&l0[(koff)];                      \
        F1 = *(const v4f*)&l1[(koff)];                      \
    } while (0)

#define GROUP8(A0, A1, F0, F1)                                              \
    do {                                                                    \
        v2f a0 = LOQ(A0), a1 = LOQ(A1);                                     \
        v2f f0 = LOQ(F0), f1 = LOQ(F1);                                     \
        c00 = WMMA_F32(a0, f0, c00);  c01 = WMMA_F32(a0, f1, c01);          \
        c10 = WMMA_F32(a1, f0, c10);  c11 = WMMA_F32(a1, f1, c11);          \
        a0 = HIQ(A0); a1 = HIQ(A1); f0 = HIQ(F0); f1 = HIQ(F1);             \
        c00 = WMMA_F32(a0, f0, c00);  c01 = WMMA_F32(a0, f1, c01);          \
        c10 = WMMA_F32(a1, f0, c10);  c11 = WMMA_F32(a1, f1, c11);          \
    } while (0)

    // --- prologue: q = K-group [0,8), p = K-group [8,16) ---
    LOAD_GROUP(qA0, qA1, qf0, qf1, 0);
    LOAD_GROUP(pA0, pA1, pf0, pf1, 8);

    #pragma clang loop unroll(disable)
    for (int k0 = 0; k0 < K - 16; k0 += 16) {   // trip count = K/16 - 1
        GROUP8(qA0, qA1, qf0, qf1);                   // compute [k0, k0+8)
        LOAD_GROUP(qA0, qA1, qf0, qf1, k0 + 16);      // refill q
        GROUP8(pA0, pA1, pf0, pf1);                   // compute [k0+8, k0+16)
        LOAD_GROUP(pA0, pA1, pf0, pf1, k0 + 24);      // refill p
    }

    // --- tail: last two K-groups ---
    GROUP8(qA0, qA1, qf0, qf1);
    GROUP8(pA0, pA1, pf0, pf1);

#undef LOAD_GROUP
#undef GROUP8

    // --- ReLU + store. C/D layout: VGPR i, half h -> M = tile + 8h + i, N = r ---
    const size_t t16 = (size_t)16 * Bn;
    float* p0 = out + (size_t)(m0 + half * 8) * Bn + (b0 + r);
    float* p1 = p0 + t16;
    #pragma unroll
    for (int i = 0; i < 8; ++i) {
        float v00 = c00[i], v01 = c01[i], v10 = c10[i], v11 = c11[i];
        p0[0]  = v00 > 0.0f ? v00 : 0.0f;
        p0[16] = v01 > 0.0f ? v01 : 0.0f;
        p1[0]  = v10 > 0.0f ? v10 : 0.0f;
        p1[16] = v11 > 0.0f ? v11 : 0.0f;
        p0 += Bn; p1 += Bn;
    }
}

// ---------------------------------------------------------------------------
// Scalar fallback GEMM (only used if sizes deviate; never hit with harness).
// ---------------------------------------------------------------------------
__global__ void gemm_relu_fallback_kernel(const float* __restrict__ W,
                                          const float* __restrict__ combined,
                                          float* __restrict__ out,
                                          int E, int K, int Bn) {
    int idx = blockIdx.x * blockDim.x + threadIdx.x;
    if (idx >= E * Bn) return;
    int e = idx / Bn, b = idx - e * Bn;
    float acc = 0.0f;
    for (int k = 0; k < K; ++k)
        acc += W[(size_t)e * K + k] * combined[(size_t)b * K + k];
    out[(size_t)e * Bn + b] = acc > 0.0f ? acc : 0.0f;
}

// ---------------------------------------------------------------------------
extern "C" void kernel_launch(void* const* d_in, const int* in_sizes, int n_in,
                              void* d_out, int out_size, void* d_ws, size_t ws_size,
                              hipStream_t stream) {
    const float* features_ue = (const float*)d_in[0];
    // d_in[1] = features_ap (unused: sample_model 'ue2ue')
    const int*   neigh_idx   = (const int*)d_in[2];
    const int*   nodes       = (const int*)d_in[3];
    const float* weight      = (const float*)d_in[4];
    float*       out         = (float*)d_out;

    const int Bn = in_sizes[3];            // 20000
    const int S  = in_sizes[2] / Bn;       // 10
    const int E  = out_size / Bn;          // 256
    const int K  = in_sizes[4] / E;        // 512 (= 2F)
    const int F  = K / 2;                  // 256

    float* combined = (float*)d_ws;        // Bn * K floats

    // Phase 1: gather + mean into workspace.
    gather_combine_kernel<<<Bn, 256, 0, stream>>>(features_ue, neigh_idx, nodes,
                                                  combined, F, S);

    // Phase 2: WMMA GEMM + ReLU (specialized K=512), scalar fallback otherwise.
    if (K == 512 && E == 256 && (Bn % 32) == 0) {
        wmma_gemm_relu_kernel<512><<<dim3(Bn / 32), 256, 0, stream>>>(
            weight, combined, out, Bn);
    } else {
        int total = E * Bn;
        gemm_relu_fallback_kernel<<<(total + 255) / 256, 256, 0, stream>>>(
            weight, combined, out, E, K, Bn);
    }
}